// QuantumLayer_26328149525139
// MI455X (gfx1250) — compile-verified
//
#include <hip/hip_runtime.h>
#include <math.h>

typedef __attribute__((ext_vector_type(2))) float v2f;
typedef __attribute__((ext_vector_type(8))) float v8f;

#define NQ     12
#define NL     6
#define DIM    4096      // 2^12 amplitudes
#define BATCH  256

// ---------------------------------------------------------------------------
// Kernel A: build the 18 fused group matrices (6 layers x 3 groups of 4 qubits).
// M[l][g] = U_{4g} (x) U_{4g+1} (x) U_{4g+2} (x) U_{4g+3}   (16x16 complex)
// Stored transposed-for-WMMA-A:  ws[base + k*16 + m] = Re M[m][k], +256 for Im.
// ---------------------------------------------------------------------------
__global__ __launch_bounds__(256) void build_group_mats(const float* __restrict__ w,
                                                        float* __restrict__ mats) {
    const int lg = blockIdx.x;           // 0..17  = l*3+g
    const int l  = lg / 3;
    const int g  = lg % 3;
    const int tid = threadIdx.x;         // 0..255
    const int k = tid >> 4;              // input index a' (column)
    const int m = tid & 15;              // output index a (row)

    float re = 1.0f, im = 0.0f;
    #pragma unroll
    for (int j = 0; j < 4; ++j) {
        const int q = 4 * g + j;
        const float phi   = w[(l * NQ + q) * 3 + 0];
        const float theta = w[(l * NQ + q) * 3 + 1];
        const float omega = w[(l * NQ + q) * 3 + 2];
        float sa, ca, sb, cb, st, ct;
        __sincosf(0.5f * (phi + omega), &sa, &ca);
        __sincosf(0.5f * (phi - omega), &sb, &cb);
        __sincosf(0.5f * theta,         &st, &ct);

        const int r = (m >> (3 - j)) & 1;
        const int c = (k >> (3 - j)) & 1;
        float er, ei;
        if (r == 0 && c == 0)      { er =  ct * ca; ei = -ct * sa; }  // conj(ep)*c
        else if (r == 0 && c == 1) { er = -st * cb; ei = -st * sb; }  // -em*s
        else if (r == 1 && c == 0) { er =  st * cb; ei = -st * sb; }  // conj(em)*s
        else                       { er =  ct * ca; ei =  ct * sa; }  // ep*c
        const float nr = re * er - im * ei;
        const float ni = re * ei + im * er;
        re = nr; im = ni;
    }
    const int base = lg * 512;
    mats[base + k * 16 + m]       = re;
    mats[base + 256 + k * 16 + m] = im;
}

// Compile-time specialized index mixing: place group-nibble a among the
// remaining 8 bits u.  G0 = bits 11..8, G1 = bits 7..4, G2 = bits 3..0.
template <int G>
__device__ __forceinline__ int mixIdx(int a, int u) {
    if constexpr (G == 0) return (a << 8) | u;
    else if constexpr (G == 1) return ((u >> 4) << 8) | (a << 4) | (u & 15);
    else return (u << 4) | a;
}

// Apply one fused 16x16 complex group matrix to the LDS state with
// V_WMMA_F32_16X16X4_F32 (complex GEMM = 4 real WMMAs per K-slice of 4).
template <int G>
__device__ __forceinline__ void applyGroup(const float* __restrict__ Mr,
                                           const float* __restrict__ Mi,
                                           float (&sRe)[DIM], float (&sIm)[DIM],
                                           int wave, int mm, int khalf, int aBase) {
    // A operand: all 4 K-slices, loaded once per group (shared by both tiles).
    v2f Ar[4], Ai[4], nAi[4];
    #pragma unroll
    for (int s = 0; s < 4; ++s) {
        const int k0 = 4 * s + khalf;
        Ar[s].x  = Mr[k0 * 16 + mm];  Ar[s].y  = Mr[(k0 + 1) * 16 + mm];
        Ai[s].x  = Mi[k0 * 16 + mm];  Ai[s].y  = Mi[(k0 + 1) * 16 + mm];
        nAi[s].x = -Ai[s].x;          nAi[s].y = -Ai[s].y;
    }
    #pragma unroll
    for (int t = 0; t < 2; ++t) {            // 2 exclusive column tiles per wave
        const int col = (wave * 2 + t) * 16 + mm;
        v8f accRe = {}; v8f accIm = {};
        #pragma unroll
        for (int s = 0; s < 4; ++s) {
            const int k0 = 4 * s + khalf;
            v2f Br, Bi;
            const int i0 = mixIdx<G>(k0,     col);
            const int i1 = mixIdx<G>(k0 + 1, col);
            Br.x = sRe[i0]; Br.y = sRe[i1];
            Bi.x = sIm[i0]; Bi.y = sIm[i1];
            // Re += Ar*Br - Ai*Bi ; Im += Ar*Bi + Ai*Br
            accRe = __builtin_amdgcn_wmma_f32_16x16x4_f32(
                false, Ar[s], false, Br, (short)0, accRe, false, false);
            accRe = __builtin_amdgcn_wmma_f32_16x16x4_f32(
                false, nAi[s], false, Bi, (short)0, accRe, false, false);
            accIm = __builtin_amdgcn_wmma_f32_16x16x4_f32(
                false, Ar[s], false, Bi, (short)0, accIm, false, false);
            accIm = __builtin_amdgcn_wmma_f32_16x16x4_f32(
                false, Ai[s], false, Br, (short)0, accIm, false, false);
        }
        #pragma unroll
        for (int r = 0; r < 8; ++r) {        // D layout -> LDS (in-place: own cols)
            const int idx = mixIdx<G>(aBase + r, col);
            sRe[idx] = accRe[r];
            sIm[idx] = accIm[r];
        }
    }
    __syncthreads();
}

// ---------------------------------------------------------------------------
// Kernel B: one workgroup (8 wave32s) per batch element; state in LDS.
// ---------------------------------------------------------------------------
__global__ __launch_bounds__(256) void qsim(const float* __restrict__ x,
                                            const float* __restrict__ mats,
                                            float* __restrict__ out) {
    __shared__ float sRe[DIM];
    __shared__ float sIm[DIM];
    __shared__ float red[12 * 256];
    __shared__ float cs[NQ], sn[NQ];

    const int tid = threadIdx.x;
    const int b   = blockIdx.x;

    if (tid < NQ) {
        float s, c;
        __sincosf(0.5f * x[b * NQ + tid], &s, &c);
        cs[tid] = c;
        sn[tid] = s;
    }
    __syncthreads();

    // --- initial product state: amp(j) = prod_q (bit?sin:cos) * (-i)^popc(j)
    #pragma unroll 1
    for (int i = 0; i < 16; ++i) {
        const int j = tid * 16 + i;
        float mag = 1.0f;
        #pragma unroll
        for (int q = 0; q < NQ; ++q) {
            const int bit = (j >> (11 - q)) & 1;
            mag *= bit ? sn[q] : cs[q];
        }
        const int pc = __popc(j) & 3;
        float re, im;
        if (pc == 0)      { re =  mag; im = 0.0f; }
        else if (pc == 1) { re = 0.0f; im = -mag; }
        else if (pc == 2) { re = -mag; im = 0.0f; }
        else              { re = 0.0f; im =  mag; }
        sRe[j] = re; sIm[j] = im;
    }
    __syncthreads();

    const int lane  = tid & 31;
    const int wave  = tid >> 5;
    const int mm    = lane & 15;            // A: row m / B: column n / D: N
    const int khalf = (lane >> 4) << 1;     // 0 or 2 (K sub-slice per lane half)
    const int aBase = (lane >> 4) << 3;     // D rows 0..7 or 8..15

    #pragma unroll 1
    for (int l = 0; l < NL; ++l) {
        const float* M0 = mats + l * 3 * 512;
        applyGroup<0>(M0,        M0 + 256,  sRe, sIm, wave, mm, khalf, aBase);
        applyGroup<1>(M0 + 512,  M0 + 768,  sRe, sIm, wave, mm, khalf, aBase);
        applyGroup<2>(M0 + 1024, M0 + 1280, sRe, sIm, wave, mm, khalf, aBase);

        // ---- CNOT ring permutation: psi'[j] = psi[src(j)] ----
        float pr[16], pi[16];
        #pragma unroll
        for (int i = 0; i < 16; ++i) {
            const int j = tid * 16 + i;
            const int lsb   = j & 1;
            const int low   = (j ^ (j >> 1)) & 0x3FF;               // bits 0..9
            const int bit10 = ((j >> 10) ^ (j >> 11) ^ lsb) & 1;
            const int bit11 = ((j >> 11) ^ lsb) & 1;
            const int src = low | (bit10 << 10) | (bit11 << 11);
            pr[i] = sRe[src]; pi[i] = sIm[src];
        }
        __syncthreads();
        #pragma unroll
        for (int i = 0; i < 16; ++i) {
            const int j = tid * 16 + i;
            sRe[j] = pr[i]; sIm[j] = pi[i];
        }
        __syncthreads();
    }

    // ---- <Z_q> = sum_j |amp(j)|^2 * (1 - 2*bit_q(j)) ----
    float acc[NQ];
    #pragma unroll
    for (int q = 0; q < NQ; ++q) acc[q] = 0.0f;
    #pragma unroll 1
    for (int i = 0; i < 16; ++i) {
        const int j = tid * 16 + i;
        const float p = sRe[j] * sRe[j] + sIm[j] * sIm[j];
        #pragma unroll
        for (int q = 0; q < NQ; ++q) {
            const int bit = (j >> (11 - q)) & 1;
            acc[q] += bit ? -p : p;
        }
    }
    #pragma unroll
    for (int q = 0; q < NQ; ++q) red[q * 256 + tid] = acc[q];
    __syncthreads();

    // deterministic binary-tree reduction over the 256 partials per qubit
    #pragma unroll 1
    for (int off = 128; off > 0; off >>= 1) {
        if (tid < off) {
            #pragma unroll
            for (int q = 0; q < NQ; ++q)
                red[q * 256 + tid] += red[q * 256 + tid + off];
        }
        __syncthreads();
    }
    if (tid < NQ) out[b * NQ + tid] = red[tid * 256];
}

// ---------------------------------------------------------------------------
extern "C" void kernel_launch(void* const* d_in, const int* in_sizes, int n_in,
                              void* d_out, int out_size, void* d_ws, size_t ws_size,
                              hipStream_t stream) {
    const float* x = (const float*)d_in[0];        // (256, 12)
    const float* w = (const float*)d_in[1];        // (6, 12, 3)
    float* out  = (float*)d_out;                   // (256, 12)
    float* mats = (float*)d_ws;                    // 18 * 512 floats = 36 KB

    build_group_mats<<<NL * 3, 256, 0, stream>>>(w, mats);
    qsim<<<BATCH, 256, 0, stream>>>(x, mats, out);
}